// FrustumToBEV_18193481466467
// MI455X (gfx1250) — compile-verified
//
#include <hip/hip_runtime.h>

typedef __attribute__((ext_vector_type(2))) float v2f;
typedef __attribute__((ext_vector_type(8))) float v8f;

// ---------------- Problem constants (from reference setup) ----------------
#define BB      2
#define NCAM    6
#define DD      48
#define HF      28
#define WF      60
#define CCH     64
#define GX      160
#define GY      160
#define GZ      2

#define ROWS_TOTAL      (BB*NCAM*DD*HF*WF)        // 967,680 rows of 64 ch
#define ROWS_PER_BLOCK  128                        // 8 waves x 16 rows
#define LDS_STRIDE      68                         // 64 + 4 pad floats -> conflict-free A reads

// =========================================================================
// Stage 1: S[row] = sum_{c=0..63} frustum[row, c]
//   - coalesced float4 global loads into padded LDS tile
//   - WMMA f32 16x16x4 with B = ones does 16 row-sums per wave
// =========================================================================
__global__ __launch_bounds__(256) void reduce_c_wmma(const float* __restrict__ fr,
                                                     float* __restrict__ S) {
    __shared__ float tile[ROWS_PER_BLOCK * LDS_STRIDE];   // 34,816 B

    const int t        = threadIdx.x;
    const int rowBase  = blockIdx.x * ROWS_PER_BLOCK;

    // ---- cooperative coalesced load: 128 rows x 64 ch = 2048 float4 ----
    const float4* src = (const float4*)(fr) + rowBase * (CCH / 4);
#pragma unroll
    for (int i = 0; i < 8; ++i) {
        int idx = t + i * 256;              // float4 index in [0,2048)
        float4 v = src[idx];
        int row = idx >> 4;                 // 16 float4 per row
        int col = (idx & 15) << 2;          // float column
        *(float4*)&tile[row * LDS_STRIDE + col] = v;   // 16B-aligned (272B rows)
    }
    __syncthreads();

    // ---- WMMA row-sum: wave w reduces rows [w*16, w*16+16) ----
    const int wave  = t >> 5;
    const int lane  = t & 31;
    const int rsel  = lane & 15;            // A-matrix row M = rsel  (doc layout)
    const int koff  = (lane >> 4) << 1;     // lanes 0-15: K={0,1}; lanes 16-31: K={2,3}
    const int wRow  = wave * 16;

    v8f acc = {0.f,0.f,0.f,0.f,0.f,0.f,0.f,0.f};
    v2f ones; ones[0] = 1.0f; ones[1] = 1.0f;

#pragma unroll
    for (int k = 0; k < 16; ++k) {          // 16 chunks of K=4 cover C=64
        const float* p = &tile[(wRow + rsel) * LDS_STRIDE + k * 4 + koff];
        v2f a = *(const v2f*)p;             // ds_load_b64, conflict-free (stride 68)
        // D = A(16x4) * ones(4x16) + acc : every column of D = row sums
        acc = __builtin_amdgcn_wmma_f32_16x16x4_f32(
                  false, a, false, ones, (short)0, acc, false, false);
    }

    // D layout: VGPR r, lane 0 -> M=r ; lane 16 -> M=8+r. Column 0 suffices.
    if ((lane & 15) == 0) {
        int outBase = rowBase + wRow + ((lane >> 4) << 3);  // rows 0-7 or 8-15
        float4 lo = {acc[0], acc[1], acc[2], acc[3]};
        float4 hi = {acc[4], acc[5], acc[6], acc[7]};
        *(float4*)(S + outBase)     = lo;
        *(float4*)(S + outBase + 4) = hi;
    }
}

// =========================================================================
// Stage 2: project voxels, trilinear-sample S, sum over cameras
//   one thread per bev element [B, Z, Y, X] = 102,400 threads
// =========================================================================
__device__ __forceinline__ float corner_sample(const float* __restrict__ S,
                                               int camBase, int w, int h, int d,
                                               float wt) {
    bool valid = (w >= 0) & (w < WF) & (h >= 0) & (h < HF) & (d >= 0) & (d < DD);
    int wc = min(max(w, 0), WF - 1);
    int hc = min(max(h, 0), HF - 1);
    int dc = min(max(d, 0), DD - 1);
    float v = S[camBase + (dc * HF + hc) * WF + wc];
    return valid ? v * wt : 0.0f;
}

__global__ __launch_bounds__(256) void splat_bev(const float* __restrict__ S,
                                                 const float* __restrict__ intr,
                                                 const float* __restrict__ l2s,
                                                 const float* __restrict__ grid,
                                                 const int* __restrict__ pImgH,
                                                 const int* __restrict__ pImgW,
                                                 float* __restrict__ out) {
    int t = blockIdx.x * 256 + threadIdx.x;           // [B,Z,Y,X] flat
    int x = t % GX;
    int y = (t / GX) % GY;
    int z = (t / (GX * GY)) % GZ;
    int b = t / (GX * GY * GZ);                       // uniform per block

    const float imgW = (float)pImgW[0];
    const float imgH = (float)pImgH[0];

    // world-space voxel (read the actual grid input to match reference bitwise)
    int gi = ((x * GY + y) * GZ + z) * 4;
    float gx = grid[gi + 0];
    float gy = grid[gi + 1];
    float gz = grid[gi + 2];

    float acc = 0.0f;

#pragma unroll
    for (int n = 0; n < NCAM; ++n) {
        const float* K = intr + (b * NCAM + n) * 9;    // 3x3
        const float* T = l2s  + (b * NCAM + n) * 16;   // 4x4
        // proj = K @ T[:3,:]  (3x4), block-uniform -> scalar loads
        float P[12];
#pragma unroll
        for (int i = 0; i < 3; ++i)
#pragma unroll
            for (int j = 0; j < 4; ++j)
                P[i * 4 + j] = K[i * 3 + 0] * T[0 * 4 + j]
                             + K[i * 3 + 1] * T[1 * 4 + j]
                             + K[i * 3 + 2] * T[2 * 4 + j];

        float u0  = P[0] * gx + P[1] * gy + P[2]  * gz + P[3];
        float v0  = P[4] * gx + P[5] * gy + P[6]  * gz + P[7];
        float dpt = P[8] * gx + P[9] * gy + P[10] * gz + P[11];

        float u = u0 / dpt;
        float v = v0 / dpt;
        float dbin = (dpt - 2.0f);                     // (depth - D_START)/D_STEP

        // normalize exactly like reference, then grid_sample unnormalize
        float gxn = 2.0f * u    / imgW  - 1.0f;
        float gyn = 2.0f * v    / imgH  - 1.0f;
        float gzn = 2.0f * dbin / 48.0f - 1.0f;
        float ix = ((gxn + 1.0f) * (float)WF - 1.0f) * 0.5f;
        float iy = ((gyn + 1.0f) * (float)HF - 1.0f) * 0.5f;
        float iz = ((gzn + 1.0f) * (float)DD - 1.0f) * 0.5f;

        float x0 = floorf(ix), y0 = floorf(iy), z0 = floorf(iz);
        float fx = ix - x0, fy = iy - y0, fz = iz - z0;
        int xi = (int)x0, yi = (int)y0, zi = (int)z0;

        int camBase = (b * NCAM + n) * (DD * HF * WF);
        float ofx = 1.0f - fx, ofy = 1.0f - fy, ofz = 1.0f - fz;

        acc += corner_sample(S, camBase, xi,     yi,     zi,     ofx * ofy * ofz);
        acc += corner_sample(S, camBase, xi + 1, yi,     zi,     fx  * ofy * ofz);
        acc += corner_sample(S, camBase, xi,     yi + 1, zi,     ofx * fy  * ofz);
        acc += corner_sample(S, camBase, xi + 1, yi + 1, zi,     fx  * fy  * ofz);
        acc += corner_sample(S, camBase, xi,     yi,     zi + 1, ofx * ofy * fz);
        acc += corner_sample(S, camBase, xi + 1, yi,     zi + 1, fx  * ofy * fz);
        acc += corner_sample(S, camBase, xi,     yi + 1, zi + 1, ofx * fy  * fz);
        acc += corner_sample(S, camBase, xi + 1, yi + 1, zi + 1, fx  * fy  * fz);
    }

    out[t] = acc;
}

// =========================================================================
extern "C" void kernel_launch(void* const* d_in, const int* in_sizes, int n_in,
                              void* d_out, int out_size, void* d_ws, size_t ws_size,
                              hipStream_t stream) {
    const float* frustum = (const float*)d_in[0];
    const float* intr    = (const float*)d_in[1];
    const float* l2s     = (const float*)d_in[2];
    const float* grid    = (const float*)d_in[3];
    const int*   imgH    = (const int*)d_in[4];
    const int*   imgW    = (const int*)d_in[5];
    float*       out     = (float*)d_out;
    float*       S       = (float*)d_ws;              // 967,680 f32 = 3.87 MB

    reduce_c_wmma<<<ROWS_TOTAL / ROWS_PER_BLOCK, 256, 0, stream>>>(frustum, S);
    splat_bev<<<(BB * GZ * GY * GX) / 256, 256, 0, stream>>>(S, intr, l2s, grid,
                                                             imgH, imgW, out);
}